// NSRLossV2_65781719105571
// MI455X (gfx1250) — compile-verified
//
#include <hip/hip_runtime.h>
#include <hip/hip_bf16.h>
#include <stdint.h>

typedef float v2f __attribute__((ext_vector_type(2)));
typedef float v8f __attribute__((ext_vector_type(8)));

// Problem sizes (fixed by the reference)
constexpr int Bsz = 512;   // batch
constexpr int Dd  = 4096;  // input dim
constexpr int Hh  = 4096;  // hidden dim
constexpr int Cc  = 5;     // classes

// GEMM tiling
constexpr int BM  = 128;
constexpr int BN  = 128;
constexpr int BK  = 16;
constexpr int LDA  = BK + 4;   // 20 floats/row (80B, 16B-aligned rows)
constexpr int LDB1 = BN + 4;   // 132 floats/row (528B, 16B-aligned rows)

// --- CDNA5 async global->LDS helpers --------------------------------------
// LDS byte offset == low 32 bits of the generic address of a __shared__ object
__device__ __forceinline__ uint32_t lds_off_of(const void* p) {
  return (uint32_t)(uintptr_t)p;
}

// GLOBAL_LOAD_ASYNC_TO_LDS_B128: each lane copies 16 bytes from the global address in vaddr to the LDS address in vdst, counted by ASYNCcnt
__device__ __forceinline__ void async_copy_b128(uint32_t lds_byte_off,
                                                const void* gaddr) {
  asm volatile("global_load_async_to_lds_b128 %0, %1, off"
               :
               : "v"(lds_byte_off), "v"(gaddr)
               : "memory");
}

__device__ __forceinline__ void wait_async0() {
#if __has_builtin(__builtin_amdgcn_s_wait_asynccnt)
  __builtin_amdgcn_s_wait_asynccnt(0);
#else
  asm volatile("s_wait_asynccnt 0x0" ::: "memory");
#endif
}

// ---------------------------------------------------------------------------
// GEMM1: H = relu(x @ W1 + b1)   [512,4096] = [512,4096] x [4096,4096]
// f32 WMMA 16x16x4; 128x128 tile/WG; 8 waves, 64x32 per wave;
// double-buffered LDS fed by async global->LDS copies.
// ---------------------------------------------------------------------------
__global__ __launch_bounds__(256) void gemm1_relu_kernel(
    const float* __restrict__ X, const float* __restrict__ W1,
    const float* __restrict__ b1, float* __restrict__ Hact) {
  __shared__ __align__(16) float As[2][BM * LDA];    // As[buf][m][k]
  __shared__ __align__(16) float Bs[2][BK * LDB1];   // Bs[buf][k][n]

  const int n0 = blockIdx.x * BN;
  const int m0 = blockIdx.y * BM;
  const int t    = threadIdx.x;
  const int lane = t & 31;
  const int wave = t >> 5;
  const int l15  = lane & 15;
  const int lh   = lane >> 4;
  const int wm = (wave >> 2) * 64;   // 0 / 64
  const int wn = (wave & 3) * 32;    // 0..96

  // per-thread staging indices (2 float4 per tile for each of A and B)
  const int am[2] = {(t + 0) >> 2, (t + 256) >> 2};
  const int ak[2] = {((t + 0) & 3) * 4, ((t + 256) & 3) * 4};
  const int bk[2] = {(t + 0) >> 5, (t + 256) >> 5};
  const int bn[2] = {((t + 0) & 31) * 4, ((t + 256) & 31) * 4};

  auto issue_tiles = [&](int kt, int buf) {
#pragma unroll
    for (int it = 0; it < 2; ++it)
      async_copy_b128(lds_off_of(&As[buf][am[it] * LDA + ak[it]]),
                      &X[(size_t)(m0 + am[it]) * Dd + kt + ak[it]]);
#pragma unroll
    for (int it = 0; it < 2; ++it)
      async_copy_b128(lds_off_of(&Bs[buf][bk[it] * LDB1 + bn[it]]),
                      &W1[(size_t)(kt + bk[it]) * Hh + n0 + bn[it]]);
  };

  v8f acc[4][2];
  const v8f vzero = {0.f, 0.f, 0.f, 0.f, 0.f, 0.f, 0.f, 0.f};
#pragma unroll
  for (int mt = 0; mt < 4; ++mt)
#pragma unroll
    for (int nt = 0; nt < 2; ++nt) acc[mt][nt] = vzero;

  issue_tiles(0, 0);
  int p = 0;
  for (int kt = 0; kt < Dd; kt += BK) {
    wait_async0();       // this wave's async copies (tile kt) landed in LDS
    __syncthreads();     // everyone's copies landed; prev compute finished
    if (kt + BK < Dd) issue_tiles(kt + BK, p ^ 1);  // overlap next tile

#pragma unroll
    for (int ks = 0; ks < 4; ++ks) {
      const int kk = ks * 4 + lh * 2;
      v2f a[4], b[2];
#pragma unroll
      for (int mt = 0; mt < 4; ++mt)
        a[mt] = *reinterpret_cast<const v2f*>(
            &As[p][(wm + mt * 16 + l15) * LDA + kk]);
#pragma unroll
      for (int nt = 0; nt < 2; ++nt) {
        const int n = wn + nt * 16 + l15;
        b[nt].x = Bs[p][kk * LDB1 + n];
        b[nt].y = Bs[p][(kk + 1) * LDB1 + n];
      }
#pragma unroll
      for (int mt = 0; mt < 4; ++mt)
#pragma unroll
        for (int nt = 0; nt < 2; ++nt)
          acc[mt][nt] = __builtin_amdgcn_wmma_f32_16x16x4_f32(
              false, a[mt], false, b[nt], (short)0, acc[mt][nt], false, false);
    }
    p ^= 1;
  }

  // Epilogue: + bias, relu, store H
  float bias[2];
#pragma unroll
  for (int nt = 0; nt < 2; ++nt) bias[nt] = b1[n0 + wn + nt * 16 + l15];
#pragma unroll
  for (int mt = 0; mt < 4; ++mt)
#pragma unroll
    for (int nt = 0; nt < 2; ++nt)
#pragma unroll
      for (int r = 0; r < 8; ++r) {
        const int m = m0 + wm + mt * 16 + lh * 8 + r;
        const int n = n0 + wn + nt * 16 + l15;
        const float v = acc[mt][nt][r] + bias[nt];
        Hact[(size_t)m * Hh + n] = v > 0.f ? v : 0.f;
      }
}

// ---------------------------------------------------------------------------
// Logits + per-sample scalars: out = H@W2 + b2; mse/margin/argmax/z_y
// ---------------------------------------------------------------------------
__global__ __launch_bounds__(256) void logits_persample_kernel(
    const float* __restrict__ Hact, const int* __restrict__ Y,
    const float* __restrict__ W2, const float* __restrict__ b2,
    float* __restrict__ zy, float* __restrict__ scal) {
  const int i = blockIdx.x;
  const int t = threadIdx.x;
  float p[Cc] = {0.f, 0.f, 0.f, 0.f, 0.f};
  for (int h = t; h < Hh; h += 256) {
    const float hv = Hact[(size_t)i * Hh + h];
#pragma unroll
    for (int c = 0; c < Cc; ++c) p[c] += hv * W2[h * Cc + c];
  }
  __shared__ float sm[256][Cc];
#pragma unroll
  for (int c = 0; c < Cc; ++c) sm[t][c] = p[c];
  __syncthreads();
  for (int s = 128; s > 0; s >>= 1) {
    if (t < s)
#pragma unroll
      for (int c = 0; c < Cc; ++c) sm[t][c] += sm[t + s][c];
    __syncthreads();
  }
  if (t == 0) {
    float o[Cc];
#pragma unroll
    for (int c = 0; c < Cc; ++c) o[c] = sm[0][c] + b2[c];
    const int yv = Y[i];
    const float z = o[yv];
    zy[i] = z;
    float mse = 0.f, marg = 0.f;
    int pred = 0;
    float mx = o[0];
#pragma unroll
    for (int c = 0; c < Cc; ++c) {
      const float tgt = (c == yv) ? 1.f : 0.f;
      const float d = o[c] - tgt;
      mse += d * d;
      if (c != yv) {
        const float mg = 1.f - z + o[c];
        if (mg > 0.f) marg += mg;
      }
      if (c > 0 && o[c] > mx) { mx = o[c]; pred = c; }
    }
    atomicAdd(&scal[0], mse);
    atomicAdd(&scal[1], marg);
    atomicAdd(&scal[2], (pred == yv) ? 1.f : 0.f);
  }
}

// ---------------------------------------------------------------------------
// GEMM2: G = A2 @ W1^T, A2[i,h] = (H[i,h]>0) * W2[h, y_i]; fused row-L1
// reduce into w_l1[i]. B tile async double-buffered; A tile data prefetched
// into registers during the previous tile's compute.
// ---------------------------------------------------------------------------
__global__ __launch_bounds__(256) void gemm2_nsr_kernel(
    const float* __restrict__ Hact, const int* __restrict__ Y,
    const float* __restrict__ W1, const float* __restrict__ W2,
    float* __restrict__ wl1) {
  __shared__ __align__(16) float As[BM * LDA];       // As[m][k]  (computed A2 tile)
  __shared__ __align__(16) float Bs[2][BN * LDA];    // Bs[buf][n][k] (W1^T)

  const int n0 = blockIdx.x * BN;   // over D
  const int m0 = blockIdx.y * BM;   // over batch
  const int t    = threadIdx.x;
  const int lane = t & 31;
  const int wave = t >> 5;
  const int l15  = lane & 15;
  const int lh   = lane >> 4;
  const int wm = (wave >> 2) * 64;
  const int wn = (wave & 3) * 32;

  const int am[2] = {(t + 0) >> 2, (t + 256) >> 2};
  const int ak[2] = {((t + 0) & 3) * 4, ((t + 256) & 3) * 4};
  const int yv[2] = {Y[m0 + am[0]], Y[m0 + am[1]]};  // K-invariant

  auto issue_B = [&](int kt, int buf) {
#pragma unroll
    for (int it = 0; it < 2; ++it) {
      const int idx = t + it * 256;
      const int n = idx >> 2, kq = (idx & 3) * 4;
      async_copy_b128(lds_off_of(&Bs[buf][n * LDA + kq]),
                      &W1[(size_t)(n0 + n) * Hh + kt + kq]);
    }
  };

  float4 hv[2];
  float w2v[2][4];
  auto load_A_regs = [&](int kt) {
#pragma unroll
    for (int it = 0; it < 2; ++it) {
      const int hg = kt + ak[it];
      hv[it] = *reinterpret_cast<const float4*>(
          &Hact[(size_t)(m0 + am[it]) * Hh + hg]);
#pragma unroll
      for (int j = 0; j < 4; ++j) w2v[it][j] = W2[(hg + j) * Cc + yv[it]];
    }
  };

  v8f acc[4][2];
  const v8f vzero = {0.f, 0.f, 0.f, 0.f, 0.f, 0.f, 0.f, 0.f};
#pragma unroll
  for (int mt = 0; mt < 4; ++mt)
#pragma unroll
    for (int nt = 0; nt < 2; ++nt) acc[mt][nt] = vzero;

  issue_B(0, 0);
  load_A_regs(0);
  int p = 0;
  for (int kt = 0; kt < Hh; kt += BK) {
    wait_async0();
    __syncthreads();  // B tile kt in Bs[p]; prev compute (reader of As) done

    // build A2 tile in LDS from prefetched registers
#pragma unroll
    for (int it = 0; it < 2; ++it) {
      float4 a;
      a.x = hv[it].x > 0.f ? w2v[it][0] : 0.f;
      a.y = hv[it].y > 0.f ? w2v[it][1] : 0.f;
      a.z = hv[it].z > 0.f ? w2v[it][2] : 0.f;
      a.w = hv[it].w > 0.f ? w2v[it][3] : 0.f;
      *reinterpret_cast<float4*>(&As[am[it] * LDA + ak[it]]) = a;
    }
    if (kt + BK < Hh) {
      issue_B(kt + BK, p ^ 1);   // overlap next B tile
      load_A_regs(kt + BK);      // overlap next A data (regs)
    }
    __syncthreads();  // A2 tile visible to all waves

#pragma unroll
    for (int ks = 0; ks < 4; ++ks) {
      const int kk = ks * 4 + lh * 2;
      v2f a[4], b[2];
#pragma unroll
      for (int mt = 0; mt < 4; ++mt)
        a[mt] = *reinterpret_cast<const v2f*>(
            &As[(wm + mt * 16 + l15) * LDA + kk]);
#pragma unroll
      for (int nt = 0; nt < 2; ++nt)
        b[nt] = *reinterpret_cast<const v2f*>(
            &Bs[p][(wn + nt * 16 + l15) * LDA + kk]);
#pragma unroll
      for (int mt = 0; mt < 4; ++mt)
#pragma unroll
        for (int nt = 0; nt < 2; ++nt)
          acc[mt][nt] = __builtin_amdgcn_wmma_f32_16x16x4_f32(
              false, a[mt], false, b[nt], (short)0, acc[mt][nt], false, false);
    }
    p ^= 1;
  }

  // Fused epilogue: w_l1[i] += sum_n |G[i,n]| over this block's 128 n's
#pragma unroll
  for (int mt = 0; mt < 4; ++mt) {
#pragma unroll
    for (int r = 0; r < 8; ++r) {
      float s = fabsf(acc[mt][0][r]) + fabsf(acc[mt][1][r]);
      s += __shfl_xor(s, 1);
      s += __shfl_xor(s, 2);
      s += __shfl_xor(s, 4);
      s += __shfl_xor(s, 8);
      if (l15 == 0) {
        const int m = m0 + wm + mt * 16 + lh * 8 + r;
        atomicAdd(&wl1[m], s);
      }
    }
  }
}

// ---------------------------------------------------------------------------
// Finalize: nsr = BETA*mean(log(w_l1*EPS/(|z_y|+1e-8)+1)); assemble loss
// ---------------------------------------------------------------------------
__global__ __launch_bounds__(512) void finalize_kernel(
    const float* __restrict__ zy, const float* __restrict__ wl1,
    const float* __restrict__ scal, float* __restrict__ out) {
  __shared__ float sm[512];
  const int t = threadIdx.x;
  const float z = zy[t];
  const float R = wl1[t] * 0.05f / (fabsf(z) + 1e-8f);
  sm[t] = logf(R + 1.f);
  __syncthreads();
  for (int s = 256; s > 0; s >>= 1) {
    if (t < s) sm[t] += sm[t + s];
    __syncthreads();
  }
  if (t == 0) {
    const float nsr  = 0.4f * sm[0] / 512.f;
    const float mse  = scal[0] / (512.f * 5.f);
    const float marg = scal[1] / 512.f;
    const float gate = scal[2] / 512.f;
    out[0] = mse + (marg + nsr) * gate;
  }
}

__global__ void zero_kernel(float* __restrict__ p, int n) {
  const int i = blockIdx.x * 256 + threadIdx.x;
  if (i < n) p[i] = 0.f;
}

// ---------------------------------------------------------------------------
extern "C" void kernel_launch(void* const* d_in, const int* in_sizes, int n_in,
                              void* d_out, int out_size, void* d_ws, size_t ws_size,
                              hipStream_t stream) {
  const float* X  = (const float*)d_in[0];
  const int*   Y  = (const int*)d_in[1];
  const float* W1 = (const float*)d_in[2];
  const float* b1 = (const float*)d_in[3];
  const float* W2 = (const float*)d_in[4];
  const float* b2 = (const float*)d_in[5];

  float* ws   = (float*)d_ws;
  float* Hact = ws;                              // 512*4096 floats (8 MB)
  float* zy   = Hact + (size_t)Bsz * Hh;         // 512
  float* wl1  = zy + Bsz;                        // 512
  float* scal = wl1 + Bsz;                       // [mse_sum, margin_sum, correct]

  zero_kernel<<<3, 256, 0, stream>>>(wl1, Bsz + 3);
  gemm1_relu_kernel<<<dim3(Hh / BN, Bsz / BM), 256, 0, stream>>>(X, W1, b1, Hact);
  logits_persample_kernel<<<Bsz, 256, 0, stream>>>(Hact, Y, W2, b2, zy, scal);
  gemm2_nsr_kernel<<<dim3(Dd / BN, Bsz / BM), 256, 0, stream>>>(Hact, Y, W1, W2, wl1);
  finalize_kernel<<<1, 512, 0, stream>>>(zy, wl1, scal, (float*)d_out);
}